// CropRoi_74483322847688
// MI455X (gfx1250) — compile-verified
//
#include <hip/hip_runtime.h>
#include <stdint.h>
#include <stddef.h>

// ---------------------------------------------------------------------------
// CropRoi (3D ROI adaptive max-pool) for MI455X / gfx1250.
//   f         : (2, 128, 24, 24, 24) fp32
//   inputs    : (2, 1, 96, 96, 96)   fp32 (shape-only, unused)
//   proposals : (64, 8) fp32  [b, score, cz, cy, cx, sz, sy, sx]
//   scale     : int (4)
//   out       : (64, 128, 7, 7, 7) fp32
// Strategy: one workgroup per (proposal, channel). Stage the clipped box
// (<= 24^3 floats = 55 KB) into LDS with CDNA5 async global->LDS loads,
// s_wait_asynccnt + barrier, then 343 threads each max-reduce one output bin
// from LDS. Bins are disjoint (except 1-elem boundary overlap), so the box is
// read from global exactly once per channel.
// ---------------------------------------------------------------------------

#define S_OUT  7
#define NPROP  64
#define NCH    128
#define DF     24                 // feature-space extent per spatial dim
#define HWF    (DF * DF)
#define OUTPN  (S_OUT * S_OUT * S_OUT)   // 343
#define BLOCK  352                // 11 wave32s; threads 0..342 each own a bin

#if __has_builtin(__builtin_amdgcn_global_load_async_to_lds_b32)
#define HAVE_ASYNC_LDS 1
#else
#define HAVE_ASYNC_LDS 0
#warning "CDNA5 probe: __builtin_amdgcn_global_load_async_to_lds_b32 NOT available on this toolchain"
#endif

#if __has_builtin(__builtin_amdgcn_s_wait_asynccnt)
#define WAIT_ASYNCCNT0() __builtin_amdgcn_s_wait_asynccnt(0)
#else
#define WAIT_ASYNCCNT0() asm volatile("s_wait_asynccnt 0" ::: "memory")
#endif

// hipcc prints addrspace(1) as `__device__ int *` in the diagnostic, so the
// async builtin takes int-typed AS(1)/AS(3) pointers.
typedef __attribute__((address_space(1))) int as1_int;
typedef __attribute__((address_space(3))) int as3_int;

__global__ __launch_bounds__(BLOCK)
void crop_roi_max_kernel(const float* __restrict__ f,
                         const float* __restrict__ props,
                         const int* __restrict__ scale_p,
                         float* __restrict__ out)
{
    __shared__ float box[DF * DF * DF];   // 55,296 B

    const int n   = blockIdx.x;   // proposal
    const int c   = blockIdx.y;   // channel
    const int tid = threadIdx.x;

    const float fscale = (float)(*scale_p);

    // --- per-proposal box bounds (computed redundantly by every thread) ----
    const float* p = props + (size_t)n * 8;
    const int b = (int)p[0];                       // trunc-toward-zero == astype(int32)

    int c0d[3], Ld[3];
#pragma unroll
    for (int d = 0; d < 3; ++d) {
        const float ctr = p[2 + d];
        const float sd  = p[5 + d];
        const float c0f = ctr - sd * 0.5f;         // matches reference exactly
        const float c1f = c0f + sd;
        int lo = (int)floorf(c0f / fscale);
        int hi = (int)ceilf(c1f / fscale);
        lo = lo > 0 ? lo : 0;
        hi = hi < DF ? hi : DF;
        c0d[d] = lo;
        int L = hi - lo;
        Ld[d] = L > 0 ? L : 0;
    }
    const int Lz = Ld[0], Ly = Ld[1], Lx = Ld[2];
    const int boxN = Lz * Ly * Lx;

    const float* fbase = f + (((size_t)(b * NCH + c) * DF + c0d[0]) * DF + c0d[1]) * DF + c0d[2];

    // --- stage box into LDS --------------------------------------------------
    for (int i = tid; i < boxN; i += BLOCK) {
        const int x = i % Lx;
        const int t = i / Lx;
        const int y = t % Ly;
        const int z = t / Ly;
        const float* g = fbase + (size_t)z * HWF + (size_t)y * DF + x;
#if HAVE_ASYNC_LDS
        // CDNA5 per-lane async copy: copies 4 bytes from global address g into box[i] in LDS, tracked by ASYNCcnt.
        __builtin_amdgcn_global_load_async_to_lds_b32(
            (as1_int*)(void*)g, (as3_int*)(void*)&box[i], /*imm offset*/0, /*cpol*/0);
#else
        box[i] = g[0];
#endif
    }
#if HAVE_ASYNC_LDS
    WAIT_ASYNCCNT0();     // each wave drains its own async loads
#endif
    __syncthreads();      // make staged LDS visible to all waves

    // --- each thread reduces one (bz,by,bx) bin ------------------------------
    if (tid < OUTPN) {
        const int bx = tid % S_OUT;
        const int t2 = tid / S_OUT;
        const int by = t2 % S_OUT;
        const int bz = t2 / S_OUT;

        // box-local adaptive bin bounds:  s = i*L/S,  e = ceil((i+1)*L/S)
        const int sz = (bz * Lz) / S_OUT, ez = ((bz + 1) * Lz + S_OUT - 1) / S_OUT;
        const int sy = (by * Ly) / S_OUT, ey = ((by + 1) * Ly + S_OUT - 1) / S_OUT;
        const int sx = (bx * Lx) / S_OUT, ex = ((bx + 1) * Lx + S_OUT - 1) / S_OUT;

        float acc = -__FLT_MAX__;    // empty bins -> finfo(f32).min, as reference
        for (int z = sz; z < ez; ++z)
            for (int y = sy; y < ey; ++y)
                for (int x = sx; x < ex; ++x)
                    acc = fmaxf(acc, box[(z * Ly + y) * Lx + x]);

        out[((size_t)(n * NCH + c)) * OUTPN + tid] = acc;
    }
}

extern "C" void kernel_launch(void* const* d_in, const int* in_sizes, int n_in,
                              void* d_out, int out_size, void* d_ws, size_t ws_size,
                              hipStream_t stream)
{
    (void)in_sizes; (void)n_in; (void)out_size; (void)d_ws; (void)ws_size;
    const float* f     = (const float*)d_in[0];
    // d_in[1] ("inputs") is shape-only; unused.
    const float* props = (const float*)d_in[2];
    const int*   scale = (const int*)d_in[3];
    float*       out   = (float*)d_out;

    dim3 grid(NPROP, NCH);
    crop_roi_max_kernel<<<grid, BLOCK, 0, stream>>>(f, props, scale, out);
}